// OctreeCrossEntropyLoss_65163243815393
// MI455X (gfx1250) — compile-verified
//
#include <hip/hip_runtime.h>
#include <math.h>

// ---------------------------------------------------------------------------
// Octree cross-entropy loss, MI455X (gfx1250, wave32).
// Level 0: memory-bound streaming pass (201 MB @ 23.3 TB/s ~ 8.6 us floor).
// Levels 1-4: gt is {0,1} so block min/max == AND/OR of level-0 block bits.
// WMMA f32 16x16x4 used to reduce (swn, sw) lane partials in one matrix op.
// ---------------------------------------------------------------------------

#define DDIM   256
#define NBLK0  4096      // 16^3 level-0 blocks
#define W_IN   16.0f
#define W_OUT  1.0f

typedef float v2f __attribute__((ext_vector_type(2)));
typedef float v8f __attribute__((ext_vector_type(8)));

// packed summary: bit0 = block min (AND), bit1 = block max (OR)
__device__ __forceinline__ unsigned mm_combine(unsigned a, unsigned b) {
    return (a & b & 1u) | ((a | b) & 2u);
}

// ---------------------------------------------------------------------------
// Kernel 1: one workgroup (256 threads = 8 waves) per 16^3 block.
// Thread t <-> (ix = t>>4, iy = t&15), covers all 16 iz (contiguous 64B runs
// in gt and in preds -> b128 loads). Produces blockLoss[bi] = sum(w*nll)/sum(w)
// and blockMM[bi] = min/max bits of gt within the block.
// ---------------------------------------------------------------------------
__global__ __launch_bounds__(256)
void octree_level0_kernel(const int*   __restrict__ gt,
                          const float* __restrict__ preds,
                          float*       __restrict__ blockLoss,
                          unsigned*    __restrict__ blockMM) {
    const int bi = blockIdx.x;
    const int t  = threadIdx.x;
    const int bx = bi >> 8, by = (bi >> 4) & 15, bz = bi & 15;
    const int ix = t >> 4,  iy = t & 15;

    // 16 consecutive gt ints for this (ix,iy) row of the block (64B aligned)
    const int gtOff = ((bx * 16 + ix) * DDIM + (by * 16 + iy)) * DDIM + bz * 16;
    const int4*   gptr  = (const int4*)(gt + gtOff);
    const float4* p0ptr = (const float4*)(preds + bi * 8192 + t * 16);
    const float4* p1ptr = (const float4*)(preds + bi * 8192 + 4096 + t * 16);

    // Issue all loads up front (12 x b128 per thread) to maximize MLP.
    int4   g4[4];
    float4 a4[4], b4[4];
#pragma unroll
    for (int j = 0; j < 4; ++j) { g4[j] = gptr[j]; a4[j] = p0ptr[j]; b4[j] = p1ptr[j]; }

    float    swn = 0.0f, sw = 0.0f;
    unsigned mn = 1u, mx = 0u;
#pragma unroll
    for (int j = 0; j < 4; ++j) {
        const int   gg[4] = { g4[j].x, g4[j].y, g4[j].z, g4[j].w };
        const float pa[4] = { a4[j].x, a4[j].y, a4[j].z, a4[j].w };
        const float pb[4] = { b4[j].x, b4[j].y, b4[j].z, b4[j].w };
#pragma unroll
        for (int e = 0; e < 4; ++e) {
            const float p0 = pa[e], p1 = pb[e];
            // 2-class logsumexp, numerically stable: m + log(1 + exp(-|d|))
            const float m   = fmaxf(p0, p1);
            const float lse = m + logf(1.0f + expf(-fabsf(p0 - p1)));
            const int   g   = gg[e];
            const float nll = lse - (g == 0 ? p0 : p1);
            const float w   = (g == 0) ? W_IN : W_OUT;
            swn += w * nll;
            sw  += w;
            mn &= (unsigned)g;
            mx |= (unsigned)g;
        }
    }
    unsigned mm = (mn & 1u) | ((mx << 1) & 2u);

    // intra-wave AND/OR combine of min/max bits (wave32)
#pragma unroll
    for (int s = 16; s >= 1; s >>= 1)
        mm = mm_combine(mm, (unsigned)__shfl_xor((int)mm, s, 32));

    __shared__ float    s_swn[256];
    __shared__ float    s_sw[256];
    __shared__ unsigned s_mm[8];
    s_swn[t] = swn;
    s_sw[t]  = sw;
    if ((t & 31) == 0) s_mm[t >> 5] = mm;
    __syncthreads();

    if (t < 32) {  // wave 0, EXEC all ones -> WMMA legal
        float x = 0.0f, y = 0.0f;
#pragma unroll
        for (int k = 0; k < 8; ++k) { x += s_swn[t + 32 * k]; y += s_sw[t + 32 * k]; }

        float tx, ty;
#if __has_builtin(__builtin_amdgcn_wmma_f32_16x16x4_f32)
        // A 16x4 f32 layout: VGPR0 = K0 (lanes0-15) / K2 (lanes16-31),
        //                    VGPR1 = K1 / K3. So with A = {x, y} per lane:
        //   A[m][0]=x_m  A[m][1]=y_m  A[m][2]=x_{m+16}  A[m][3]=y_{m+16}
        // B 4x16: col0 = [1,0,1,0]^T (sums x), col1 = [0,1,0,1]^T (sums y).
        v2f A; A.x = x; A.y = y;
        v2f B; B.x = ((t & 15) == 0) ? 1.0f : 0.0f;   // rows K0,K2
               B.y = ((t & 15) == 1) ? 1.0f : 0.0f;   // rows K1,K3
        v8f C = {};
        v8f Dm = __builtin_amdgcn_wmma_f32_16x16x4_f32(
            /*neg_a=*/false, A, /*neg_b=*/false, B,
            /*c_mod=*/(short)0, C, /*reuse_a=*/false, /*reuse_b=*/false);
        // D[m][n]: VGPR r holds M=r (lanes0-15) / M=r+8 (lanes16-31), N=lane%16.
        // Column 0 lives in lanes 0 & 16 (x-sums over M), column 1 in lanes 1 & 17.
        float s = Dm[0] + Dm[1] + Dm[2] + Dm[3] + Dm[4] + Dm[5] + Dm[6] + Dm[7];
        tx = __shfl(s, 0, 32) + __shfl(s, 16, 32);
        ty = __shfl(s, 1, 32) + __shfl(s, 17, 32);
#else
        // fallback: plain wave32 butterfly
#pragma unroll
        for (int s = 16; s >= 1; s >>= 1) {
            x += __shfl_xor(x, s, 32);
            y += __shfl_xor(y, s, 32);
        }
        tx = x; ty = y;
#endif
        if (t == 0) {
            unsigned m8 = s_mm[0];
#pragma unroll
            for (int k = 1; k < 8; ++k) m8 = mm_combine(m8, s_mm[k]);
            blockLoss[bi] = tx / ty;
            blockMM[bi]   = m8;
        }
    }
}

// ---------------------------------------------------------------------------
// Kernel 2: single 512-thread workgroup. Reduces 4096 block losses and walks
// the AND/OR hierarchy for levels 1..4, accumulating log(preds_single[row][g]).
// Level sizes: 512 (off 0), 64 (off 512), 8 (off 576), 1 (off 584).
// ---------------------------------------------------------------------------
__global__ __launch_bounds__(512)
void octree_levels_kernel(const float*    __restrict__ blockLoss,
                          const unsigned* __restrict__ blockMM,
                          const float*    __restrict__ preds_single,
                          float*          __restrict__ out) {
    const int t = threadIdx.x;
    __shared__ float    s_red[512];
    __shared__ unsigned s_mm1[512];
    __shared__ unsigned s_mm2[64];
    __shared__ unsigned s_mm3[8];

    float acc = 0.0f;
#pragma unroll
    for (int k = 0; k < 8; ++k) acc += blockLoss[t + 512 * k];

    // level 1: 8^3 blocks, children on the 16^3 level-0 grid
    {
        const int X = t >> 6, Y = (t >> 3) & 7, Z = t & 7;
        unsigned mn = 1u, mx = 0u;
#pragma unroll
        for (int d = 0; d < 8; ++d) {
            const int dx = d >> 2, dy = (d >> 1) & 1, dz = d & 1;
            const unsigned c = blockMM[(2 * X + dx) * 256 + (2 * Y + dy) * 16 + (2 * Z + dz)];
            mn &= c & 1u;
            mx |= (c >> 1) & 1u;
        }
        const int g = (mn != mx) ? 2 : (int)mx;
        acc += logf(preds_single[t * 3 + g]);
        s_mm1[t] = mn | (mx << 1);
    }
    __syncthreads();

    // level 2: 4^3, offset 512
    if (t < 64) {
        const int X = t >> 4, Y = (t >> 2) & 3, Z = t & 3;
        unsigned mn = 1u, mx = 0u;
#pragma unroll
        for (int d = 0; d < 8; ++d) {
            const int dx = d >> 2, dy = (d >> 1) & 1, dz = d & 1;
            const unsigned c = s_mm1[(2 * X + dx) * 64 + (2 * Y + dy) * 8 + (2 * Z + dz)];
            mn &= c & 1u;
            mx |= (c >> 1) & 1u;
        }
        const int g = (mn != mx) ? 2 : (int)mx;
        acc += logf(preds_single[(512 + t) * 3 + g]);
        s_mm2[t] = mn | (mx << 1);
    }
    __syncthreads();

    // level 3: 2^3, offset 576
    if (t < 8) {
        const int X = t >> 2, Y = (t >> 1) & 1, Z = t & 1;
        unsigned mn = 1u, mx = 0u;
#pragma unroll
        for (int d = 0; d < 8; ++d) {
            const int dx = d >> 2, dy = (d >> 1) & 1, dz = d & 1;
            const unsigned c = s_mm2[(2 * X + dx) * 16 + (2 * Y + dy) * 4 + (2 * Z + dz)];
            mn &= c & 1u;
            mx |= (c >> 1) & 1u;
        }
        const int g = (mn != mx) ? 2 : (int)mx;
        acc += logf(preds_single[(576 + t) * 3 + g]);
        s_mm3[t] = mn | (mx << 1);
    }
    __syncthreads();

    // level 4: 1 block, offset 584
    if (t == 0) {
        unsigned mn = 1u, mx = 0u;
#pragma unroll
        for (int d = 0; d < 8; ++d) {
            const unsigned c = s_mm3[d];
            mn &= c & 1u;
            mx |= (c >> 1) & 1u;
        }
        const int g = (mn != mx) ? 2 : (int)mx;
        acc += logf(preds_single[584 * 3 + g]);
    }

    // final tree reduction of 512 per-thread accumulators
    s_red[t] = acc;
    __syncthreads();
#pragma unroll
    for (int s = 256; s >= 1; s >>= 1) {
        if (t < s) s_red[t] += s_red[t + s];
        __syncthreads();
    }
    if (t == 0) out[0] = s_red[0];
}

// ---------------------------------------------------------------------------
extern "C" void kernel_launch(void* const* d_in, const int* in_sizes, int n_in,
                              void* d_out, int out_size, void* d_ws, size_t ws_size,
                              hipStream_t stream) {
    const int*   gt           = (const int*)d_in[0];     // (1,256,256,256) int32
    const float* preds_full   = (const float*)d_in[1];   // (4096,2,16,16,16) f32
    const float* preds_single = (const float*)d_in[2];   // (585,3) f32

    float*    blockLoss = (float*)d_ws;                            // 4096 f32
    unsigned* blockMM   = (unsigned*)((char*)d_ws + NBLK0 * 4);    // 4096 u32

    octree_level0_kernel<<<NBLK0, 256, 0, stream>>>(gt, preds_full, blockLoss, blockMM);
    octree_levels_kernel<<<1, 512, 0, stream>>>(blockLoss, blockMM, preds_single,
                                                (float*)d_out);
}